// QuanvolutionClassifier_65481071404724
// MI455X (gfx1250) — compile-verified
//
#include <hip/hip_runtime.h>
#include <math.h>

typedef __attribute__((ext_vector_type(2))) float v2f;
typedef __attribute__((ext_vector_type(8))) float v8f;

#define NPATCH 196
#define KDIM   784
#define HDIM   256
#define NCLS   10

// ---------------------------------------------------------------------------
// Kernel 1: exact 4-qubit quanvolution feature map. One thread per patch.
// State = 16 real fp32 amplitudes, fully unrolled -> all in VGPRs.
// ---------------------------------------------------------------------------
__global__ void quanv_kernel(const float* __restrict__ x,
                             const float* __restrict__ var_angles,
                             float* __restrict__ feats, int total) {
  int idx = blockIdx.x * blockDim.x + threadIdx.x;
  if (idx >= total) return;
  int b = idx / NPATCH;
  int p = idx - b * NPATCH;
  int pr = p / 14, pc = p - pr * 14;
  const float* img = x + b * 784 + (pr * 2) * 28 + pc * 2;
  float t0 = img[0], t1 = img[1], t2 = img[28], t3 = img[29];

  // encoder: product state  amp = ⊗_w [cos(t_w/2), sin(t_w/2)]
  float f00 = cosf(0.5f * t0), f01 = sinf(0.5f * t0);
  float f10 = cosf(0.5f * t1), f11 = sinf(0.5f * t1);
  float f20 = cosf(0.5f * t2), f21 = sinf(0.5f * t2);
  float f30 = cosf(0.5f * t3), f31 = sinf(0.5f * t3);

  float amp[16];
#pragma unroll
  for (int i = 0; i < 16; ++i) {
    float a = ((i >> 3) & 1) ? f01 : f00;
    float c = ((i >> 2) & 1) ? f11 : f10;
    float d = ((i >> 1) & 1) ? f21 : f20;
    float e = (i & 1) ? f31 : f30;
    amp[i] = a * c * d * e;
  }

  // variational layers: RY(var) on each wire, then CNOT ring 0->1->2->3->0
#pragma unroll
  for (int d = 0; d < 2; ++d) {
#pragma unroll
    for (int w = 0; w < 4; ++w) {
      float th = var_angles[d * 4 + w];
      float c = cosf(0.5f * th), s = sinf(0.5f * th);
      const int mask = 1 << (3 - w);
#pragma unroll
      for (int i = 0; i < 16; ++i) {
        if (!(i & mask)) {
          float a0 = amp[i], a1 = amp[i | mask];
          amp[i]        = c * a0 - s * a1;
          amp[i | mask] = s * a0 + c * a1;
        }
      }
    }
#pragma unroll
    for (int cw = 0; cw < 4; ++cw) {
      const int tw = (cw + 1) & 3;
      const int cm = 1 << (3 - cw), tm = 1 << (3 - tw);
#pragma unroll
      for (int i = 0; i < 16; ++i) {
        if ((i & cm) && !(i & tm)) {
          float tmpv = amp[i];
          amp[i]      = amp[i | tm];
          amp[i | tm] = tmpv;
        }
      }
    }
  }

  float prb[16];
#pragma unroll
  for (int i = 0; i < 16; ++i) prb[i] = amp[i] * amp[i];

  float* o = feats + idx * 4;  // feats[b, p*4 + w]
#pragma unroll
  for (int w = 0; w < 4; ++w) {
    const int mask = 1 << (3 - w);
    float fs = 0.f;
#pragma unroll
    for (int i = 0; i < 16; ++i) fs += (i & mask) ? -prb[i] : prb[i];
    o[w] = fs;
  }
}

// ---------------------------------------------------------------------------
// Kernel 2: h = relu(feats @ W1^T + b1) via V_WMMA_F32_16X16X4_F32.
// One wave computes one 16x16 tile of h. 8 waves / block, waves in a block
// share the same M tile (A-row reuse in L0/L2).
// A-frag: lane m=lane&15, k-half = lane>>4 -> float2 load of feats row.
// B-frag: B[k][n] = W1[n][k]  -> same float2 pattern on W1 rows.
// ---------------------------------------------------------------------------
__global__ void gemm1_relu_kernel(const float* __restrict__ feats,
                                  const float* __restrict__ W1,
                                  const float* __restrict__ b1,
                                  float* __restrict__ h, int Mtiles) {
  const int wave  = blockIdx.x * 8 + (threadIdx.x >> 5);
  const int lane  = threadIdx.x & 31;
  const int tileN = wave & 15;   // 256/16 = 16 N tiles
  const int tileM = wave >> 4;
  if (tileM >= Mtiles) return;   // wave-uniform guard: EXEC all-1s inside

  const int mn = lane & 15;            // row of A / col of B for this lane
  const int kh = (lane >> 4) << 1;     // K sub-offset: 0 or 2
  const float* aptr = feats + (size_t)(tileM * 16 + mn) * KDIM + kh;
  const float* bptr = W1    + (size_t)(tileN * 16 + mn) * KDIM + kh;

  v8f acc = {};
#pragma unroll 4
  for (int k = 0; k < KDIM; k += 4) {
    v2f a = *(const v2f*)(aptr + k);
    v2f b = *(const v2f*)(bptr + k);
    acc = __builtin_amdgcn_wmma_f32_16x16x4_f32(
        /*neg_a=*/false, a, /*neg_b=*/false, b,
        /*c_mod=*/(short)0, acc, /*reuse_a=*/false, /*reuse_b=*/false);
  }

  const int n    = tileN * 16 + mn;
  const float bi = b1[n];
  const int rowB = tileM * 16 + ((lane >> 4) << 3);  // D: vgpr r -> M = r + 8*(lane>=16)
#pragma unroll
  for (int r = 0; r < 8; ++r) {
    float v = acc[r] + bi;
    h[(size_t)(rowB + r) * HDIM + n] = fmaxf(v, 0.f);
  }
}

// ---------------------------------------------------------------------------
// Kernel 3: logits = h @ W2^T + b2 ; out = log_softmax(logits).
// One wave (32 lanes) per batch row; butterfly shfl_xor reduction leaves all
// 10 logits resident in every lane.
// ---------------------------------------------------------------------------
__global__ void head_kernel(const float* __restrict__ h,
                            const float* __restrict__ W2,
                            const float* __restrict__ b2,
                            float* __restrict__ out, int B) {
  const int lane = threadIdx.x & 31;
  const int row  = blockIdx.x * 8 + (threadIdx.x >> 5);
  if (row >= B) return;

  float acc[NCLS];
#pragma unroll
  for (int c = 0; c < NCLS; ++c) acc[c] = 0.f;

#pragma unroll
  for (int kk = 0; kk < 8; ++kk) {
    const int k = kk * 32 + lane;
    const float hv = h[(size_t)row * HDIM + k];
#pragma unroll
    for (int c = 0; c < NCLS; ++c)
      acc[c] = fmaf(hv, W2[c * HDIM + k], acc[c]);
  }

#pragma unroll
  for (int c = 0; c < NCLS; ++c) {
#pragma unroll
    for (int off = 16; off > 0; off >>= 1)
      acc[c] += __shfl_xor(acc[c], off, 32);
    acc[c] += b2[c];
  }

  float mx = acc[0];
#pragma unroll
  for (int c = 1; c < NCLS; ++c) mx = fmaxf(mx, acc[c]);
  float se = 0.f;
#pragma unroll
  for (int c = 0; c < NCLS; ++c) se += expf(acc[c] - mx);
  const float lse = mx + logf(se);

  if (lane < NCLS) {
    float myv = 0.f;
#pragma unroll
    for (int c = 0; c < NCLS; ++c)
      if (lane == c) myv = acc[c];
    out[(size_t)row * NCLS + lane] = myv - lse;
  }
}

// ---------------------------------------------------------------------------
extern "C" void kernel_launch(void* const* d_in, const int* in_sizes, int n_in,
                              void* d_out, int out_size, void* d_ws, size_t ws_size,
                              hipStream_t stream) {
  const float* x  = (const float*)d_in[0];  // [B,1,28,28]
  const float* va = (const float*)d_in[1];  // [2,4]
  const float* W1 = (const float*)d_in[2];  // [256,784]
  const float* b1 = (const float*)d_in[3];  // [256]
  const float* W2 = (const float*)d_in[4];  // [10,256]
  const float* b2 = (const float*)d_in[5];  // [10]
  float* out = (float*)d_out;               // [B,10]

  const int B = in_sizes[0] / 784;

  float* feats = (float*)d_ws;              // [B,784]
  float* h     = feats + (size_t)B * KDIM;  // [B,256]

  // 1) quanv features
  {
    int total  = B * NPATCH;
    int blocks = (total + 255) / 256;
    quanv_kernel<<<blocks, 256, 0, stream>>>(x, va, feats, total);
  }
  // 2) GEMM1 + bias + relu  (WMMA f32 16x16x4)
  {
    int Mtiles = B / 16;               // B = 4096 -> 256
    int waves  = Mtiles * 16;          // 16 N tiles each
    int blocks = (waves + 7) / 8;      // 8 waves / block
    gemm1_relu_kernel<<<blocks, 256, 0, stream>>>(feats, W1, b1, h, Mtiles);
  }
  // 3) head GEMV + log_softmax
  {
    int blocks = (B + 7) / 8;
    head_kernel<<<blocks, 256, 0, stream>>>(h, W2, b2, out, B);
  }
}